// DemonsOrientation_46308337385687
// MI455X (gfx1250) — compile-verified
//
#include <hip/hip_runtime.h>
#include <math.h>

// Volume: (1, 1, 160, 192, 160), flow: (1, 3, 160, 192, 160), output: scalar f32.
constexpr int kD = 160, kH = 192, kW = 160;
constexpr int TD = 4, THh = 8, TWw = 32;          // output tile per block
constexpr int HS = TWw + 2;                       // 34: LDS row stride (floats)
constexpr int PS = (THh + 2) * HS;                // 340: LDS plane stride
constexpr int TILE = (TD + 2) * PS;               // 2040: halo tile elements
constexpr int GX = kW / TWw, GY = kH / THh, GZ = kD / TD;   // 5 x 24 x 40
constexpr int NBLOCKS = GX * GY * GZ;             // 4800
constexpr float INV_N = 1.0f / (float)(kD * kH * kW);

__device__ __forceinline__ void async_load_f32_to_lds(const float* gptr, unsigned lds_off) {
    unsigned long long ga = (unsigned long long)(uintptr_t)gptr;
    // CDNA5 async DMA: global -> LDS, per-lane LDS address in VDST VGPR, tracked by ASYNCcnt.
    asm volatile("global_load_async_to_lds_b32 %0, %1, off"
                 :: "v"(lds_off), "v"(ga)
                 : "memory");
}

__global__ __launch_bounds__(256) void demons_ori_main(
    const float* __restrict__ M, const float* __restrict__ S,
    const float* __restrict__ flow, float* __restrict__ partial)
{
    __shared__ float sS[TILE];
    __shared__ float sM[TILE];
    __shared__ float wsum[8];

    const int tid = threadIdx.x;
    const int bw = blockIdx.x, bh = blockIdx.y, bd = blockIdx.z;
    const int gw0 = bw * TWw - 1, gh0 = bh * THh - 1, gd0 = bd * TD - 1;

    // ---- cooperative halo fill: async-DMA in-range elements, DS-store zeros for OOB ----
    for (int i = tid; i < TILE; i += 256) {
        const int lz = i / PS;  const int r  = i - lz * PS;
        const int ly = r / HS;  const int lx = r - ly * HS;
        const int gd = gd0 + lz, gh = gh0 + ly, gw = gw0 + lx;
        const bool in = ((unsigned)gd < (unsigned)kD) &
                        ((unsigned)gh < (unsigned)kH) &
                        ((unsigned)gw < (unsigned)kW);
        if (in) {
            const size_t goff = ((size_t)gd * kH + gh) * kW + gw;
            async_load_f32_to_lds(S + goff, (unsigned)(uintptr_t)&sS[i]);
            async_load_f32_to_lds(M + goff, (unsigned)(uintptr_t)&sM[i]);
        } else {
            sS[i] = 0.0f;           // zero padding (conv padding=1)
            sM[i] = 0.0f;
        }
    }
    asm volatile("s_wait_asynccnt 0x0" ::: "memory");   // drain this wave's async DMA
    __syncthreads();                                    // all waves' LDS writes visible

    // ---- per-plane Sobel aggregates (kernels factor: kx=a(h)s(w), ky=a(w)s(h), kz=s(d)a(h)) ----
    const int lw = tid & 31;        // 0..31 -> w within tile
    const int lh = tid >> 5;        // 0..7  -> h within tile
    float XS[6], YS[6], ZS[6], CS[6];
    float XM[6], YM[6], ZM[6], CM[6];
#pragma unroll
    for (int p = 0; p < 6; ++p) {
        const int b = p * PS + lh * HS + lw;   // (h-1, w-1) corner of 3x3 window
        float s00 = sS[b],          s01 = sS[b+1],          s02 = sS[b+2];
        float s10 = sS[b+HS],       s11 = sS[b+HS+1],       s12 = sS[b+HS+2];
        float s20 = sS[b+2*HS],     s21 = sS[b+2*HS+1],     s22 = sS[b+2*HS+2];
        XS[p] = (s02 - s00) + 2.0f*(s12 - s10) + (s22 - s20);
        YS[p] = (s20 - s00) + 2.0f*(s21 - s01) + (s22 - s02);
        ZS[p] = (s00 + s01 + s02) + 2.0f*(s10 + s11 + s12) + (s20 + s21 + s22);
        CS[p] = s11;
        float m00 = sM[b],          m01 = sM[b+1],          m02 = sM[b+2];
        float m10 = sM[b+HS],       m11 = sM[b+HS+1],       m12 = sM[b+HS+2];
        float m20 = sM[b+2*HS],     m21 = sM[b+2*HS+1],     m22 = sM[b+2*HS+2];
        XM[p] = (m02 - m00) + 2.0f*(m12 - m10) + (m22 - m20);
        YM[p] = (m20 - m00) + 2.0f*(m21 - m01) + (m22 - m02);
        ZM[p] = (m00 + m01 + m02) + 2.0f*(m10 + m11 + m12) + (m20 + m21 + m22);
        CM[p] = m11;
    }

    // ---- per-voxel demons orientation error vs flow ----
    float acc = 0.0f;
    const int gh = gh0 + 1 + lh;
    const int gw = gw0 + 1 + lw;
    const size_t plane = (size_t)kD * kH * kW;
#pragma unroll
    for (int ld = 0; ld < TD; ++ld) {
        float Sx = XS[ld] + XS[ld+1] + XS[ld+2];
        float Sy = YS[ld] + YS[ld+1] + YS[ld+2];
        float Sz = ZS[ld+2] - ZS[ld];
        float Mx = XM[ld] + XM[ld+1] + XM[ld+2];
        float My = YM[ld] + YM[ld+1] + YM[ld+2];
        float Mz = ZM[ld+2] - ZM[ld];
        float Id = CM[ld+1] - CS[ld+1];          // M - S
        float i2 = Id * Id;                       // ALPHA = 1
        float rS = 1.0f / (Sx*Sx + Sy*Sy + Sz*Sz + i2 + 1e-10f);
        float rM = 1.0f / (Mx*Mx + My*My + Mz*Mz + i2 + 1e-10f);
        float Ux = Id * (Sx*rS + Mx*rM);
        float Uy = Id * (Sy*rS + My*rM);
        float Uz = Id * (Sz*rS + Mz*rM);
        float ruz = 1.0f / (Uz + 1e-10f);
        float dxz = atanf(Ux * ruz);
        float dyz = atanf(Uy * ruz);

        const int gd = gd0 + 1 + ld;
        const size_t off = ((size_t)gd * kH + gh) * kW + gw;
        float fx = __builtin_nontemporal_load(flow + off);              // streaming: NT hint
        float fy = __builtin_nontemporal_load(flow + plane + off);
        float fz = __builtin_nontemporal_load(flow + 2 * plane + off);
        float rfz = 1.0f / (fz + 1e-10f);
        float exz = atanf(fx * rfz) - dxz;
        float eyz = atanf(fy * rfz) - dyz;
        acc += exz * exz + eyz * eyz;
    }

    // ---- deterministic block reduction (wave32 shuffle tree + LDS) ----
#pragma unroll
    for (int o = 16; o > 0; o >>= 1) acc += __shfl_xor(acc, o, 32);
    if ((tid & 31) == 0) wsum[tid >> 5] = acc;
    __syncthreads();
    if (tid == 0) {
        float t = 0.0f;
#pragma unroll
        for (int i = 0; i < 8; ++i) t += wsum[i];
        partial[blockIdx.x + GX * (blockIdx.y + GY * (size_t)blockIdx.z)] = t;
    }
}

__global__ __launch_bounds__(256) void demons_ori_reduce(
    const float* __restrict__ partial, float* __restrict__ out)
{
    __shared__ float wsum[8];
    float a = 0.0f;
    for (int i = threadIdx.x; i < NBLOCKS; i += 256) a += partial[i];
#pragma unroll
    for (int o = 16; o > 0; o >>= 1) a += __shfl_xor(a, o, 32);
    if ((threadIdx.x & 31) == 0) wsum[threadIdx.x >> 5] = a;
    __syncthreads();
    if (threadIdx.x == 0) {
        float t = 0.0f;
#pragma unroll
        for (int i = 0; i < 8; ++i) t += wsum[i];
        out[0] = t * INV_N;   // mean1 + mean2 == (sum1 + sum2) / N
    }
}

extern "C" void kernel_launch(void* const* d_in, const int* in_sizes, int n_in,
                              void* d_out, int out_size, void* d_ws, size_t ws_size,
                              hipStream_t stream) {
    const float* M    = (const float*)d_in[0];
    const float* S    = (const float*)d_in[1];
    const float* flow = (const float*)d_in[2];
    float* partial    = (float*)d_ws;          // NBLOCKS * 4 B = 19.2 KB of scratch
    (void)in_sizes; (void)n_in; (void)out_size; (void)ws_size;

    dim3 grid(GX, GY, GZ);
    demons_ori_main<<<grid, 256, 0, stream>>>(M, S, flow, partial);
    demons_ori_reduce<<<1, 256, 0, stream>>>(partial, (float*)d_out);
}